// LinearSpline_42451456754185
// MI455X (gfx1250) — compile-verified
//
#include <hip/hip_runtime.h>

// LinearSpline (deep-spline) activation for MI455X / gfx1250.
//
// Roofline: 33.55M fp32 elements in+out = 256 MB @ 23.3 TB/s ~= 11 us.
// Pure streaming elementwise gather+lerp -> bandwidth bound; WMMA has no
// role here. We use wave32 codegen, 128-bit NT global loads/stores, a
// conflict-free 51-entry LDS gather table, and async global->LDS staging
// of the coefficient table (ASYNCcnt path, builtin confirmed present).

typedef float v4f __attribute__((ext_vector_type(4)));

#define NCH        64
#define SZ         51
#define HALF       25         // SIZE // 2
#define GRIDW      0.04f      // 2*RANGE/(SIZE-1)
#define INV_GRID   25.0f      // 1/GRID
#define CLAMP_LO  (-25.0f)    // -(SIZE//2)        (in GRID units)
#define CLAMP_HI   24.0f      //  (SIZE//2 - 1)    (in GRID units)

#define ELEMS_PER_BLOCK 16384 // one full HxW plane -> channel is block-uniform
#define THREADS         256

typedef __attribute__((address_space(1))) int gas_int;   // global
typedef __attribute__((address_space(3))) int las_int;   // LDS

__global__ __launch_bounds__(THREADS)
void LinearSpline_42451456754185_kernel(const float* __restrict__ x,
                                        const float* __restrict__ coeff,
                                        const float* __restrict__ scal,
                                        float* __restrict__ out,
                                        long long n)
{
    __shared__ float craw[SZ];      // raw coefficients for this channel
    __shared__ float cv[SZ + 1];    // Lipschitz-rebuilt coefficients

    const int t = threadIdx.x;
    const long long blockBase = (long long)blockIdx.x * ELEMS_PER_BLOCK;
    const int c = (int)(blockIdx.x & (NCH - 1));   // 16384 = 2^14 plane size

    // ---- stage raw coefficients for channel c into LDS ------------------
    const float* csrc = coeff + c * SZ;
#if __has_builtin(__builtin_amdgcn_global_load_async_to_lds_b32)
    if (t < SZ) {
        __builtin_amdgcn_global_load_async_to_lds_b32(
            (gas_int*)(void*)(csrc + t),
            (las_int*)(void*)&craw[t],
            /*offset=*/0, /*cpol=*/0);
    }
  #if __has_builtin(__builtin_amdgcn_s_wait_asynccnt)
    __builtin_amdgcn_s_wait_asynccnt(0);
  #else
    asm volatile("s_wait_asynccnt 0" ::: "memory");
  #endif
#else
    if (t < SZ) craw[t] = csrc[t];
#endif
    __syncthreads();

    // ---- Lipschitz re-parameterization: clip slopes, cumsum, center -----
    // v_t: t==0 -> 0 ; t in [1,50] -> clamp(raw[t]-raw[t-1], 0, GRID)
    float v = 0.0f;
    if (t >= 1 && t < SZ)
        v = fminf(fmaxf(craw[t] - craw[t - 1], 0.0f), GRIDW);
    if (t < SZ) cv[t] = v;
    __syncthreads();

    // Hillis-Steele inclusive scan over 51 entries
    #pragma unroll
    for (int off = 1; off < SZ; off <<= 1) {
        float add = 0.0f;
        if (t >= off && t < SZ) add = cv[t - off];
        __syncthreads();
        if (t < SZ) cv[t] += add;
        __syncthreads();
    }
    const float mid = cv[HALF];
    __syncthreads();
    if (t < SZ) cv[t] -= mid;
    __syncthreads();

    // ---- main streaming loop: float4 per thread, 4 iterations -----------
    const float s     = scal[c];       // block-uniform -> scalar load
    const float inv_s = 1.0f / s;

    const float* __restrict__ xin  = x   + blockBase;
    float*       __restrict__ yout = out + blockBase;

    #pragma unroll
    for (int it = 0; it < 4; ++it) {
        const long long off = (long long)it * (THREADS * 4) + t * 4;
        if (blockBase + off + 3 >= n) continue;

        v4f xv = __builtin_nontemporal_load((const v4f*)(xin + off));
        v4f r;
        #pragma unroll
        for (int k = 0; k < 4; ++k) {
            const float xs = xv[k] * s;
            const float tt = xs * INV_GRID;                       // xq / GRID
            const float fl = floorf(fminf(fmaxf(tt, CLAMP_LO), CLAMP_HI));
            const float fr = tt - fl;                             // unclamped, as in ref
            const int  idx = HALF + (int)fl;                      // in [0, 49]
            const float c0 = cv[idx];
            const float c1 = cv[idx + 1];
            r[k] = (c1 * fr + c0 * (1.0f - fr)) * inv_s;
        }
        __builtin_nontemporal_store(r, (v4f*)(yout + off));
    }
}

extern "C" void kernel_launch(void* const* d_in, const int* in_sizes, int n_in,
                              void* d_out, int out_size, void* d_ws, size_t ws_size,
                              hipStream_t stream) {
    const float* x     = (const float*)d_in[0];   // [32,64,128,128] fp32
    const float* coeff = (const float*)d_in[1];   // [64*51] fp32
    const float* scal  = (const float*)d_in[2];   // [64] fp32
    float* out = (float*)d_out;

    const long long n = (long long)out_size;      // 33,554,432
    const int blocks  = (int)((n + ELEMS_PER_BLOCK - 1) / ELEMS_PER_BLOCK);

    LinearSpline_42451456754185_kernel<<<blocks, THREADS, 0, stream>>>(
        x, coeff, scal, out, n);
}